// Resample2d2_47425028883045
// MI455X (gfx1250) — compile-verified
//
#include <hip/hip_runtime.h>

// Problem constants (from reference): B=8, C=3, H=512, W=1024 — B, H, W are powers of two.
#define BB   8
#define CC   3
#define HH   512
#define WW   1024
#define HWP  (HH * WW)          // 524288  = 2^19
#define NPIX (BB * HWP)         // 4194304 = 2^22
#define HW_SHIFT 19
#define W_SHIFT  10
#define HW_MASK  (HWP - 1)
#define W_MASK   (WW - 1)
#define SAME_RANGE 0.2f

// ---------------------------------------------------------------------------
// Pass 0: zero the planar output (3*N floats), init z-buffer to +inf bits,
// zero the per-target counts. 16-byte vector stores -> global_store_b128.
// N/4 and 3N/4 are exact (N = 2^22), so no tails.
// ---------------------------------------------------------------------------
__global__ void rs2d_init_kernel(float4* __restrict__ out4,
                                 uint4*  __restrict__ dlut4,
                                 float4* __restrict__ cnt4) {
    const int i = blockIdx.x * blockDim.x + threadIdx.x;
    const int N4 = NPIX / 4;
    if (i < CC * N4) {
        out4[i] = make_float4(0.0f, 0.0f, 0.0f, 0.0f);
    }
    if (i < N4) {
        dlut4[i] = make_uint4(0x7F800000u, 0x7F800000u, 0x7F800000u, 0x7F800000u);
        cnt4[i]  = make_float4(0.0f, 0.0f, 0.0f, 0.0f);
    }
}

// ---------------------------------------------------------------------------
// Shared address computation: source pixel i -> clamped, round-to-nearest-even
// warped target pixel. Matches jnp.clip + jnp.round (banker's rounding) via
// __float2int_rn (round-to-nearest-even in default FP mode).
// ---------------------------------------------------------------------------
__device__ __forceinline__ int warp_target(const float* __restrict__ flow,
                                           int b, int rem) {
    const int y = rem >> W_SHIFT;
    const int x = rem & W_MASK;
    const size_t fbase = (size_t)b * (size_t)(2 * HWP) + (size_t)rem;
    const float fx = flow[fbase];
    const float fy = flow[fbase + HWP];
    const float px = fminf(fmaxf((float)x + fx, 0.0f), (float)(WW - 1));
    const float py = fminf(fmaxf((float)y + fy, 0.0f), (float)(HH - 1));
    const int xi = __float2int_rn(px);
    const int yi = __float2int_rn(py);
    return (yi << W_SHIFT) + xi;   // target offset within this batch's plane
}

// ---------------------------------------------------------------------------
// Pass 1: per-target z-buffer min via uint atomicMin on the depth bit pattern
// (valid ordering because all depths are in [0,1): positive floats; +inf
// sentinel dominates). Lowers to global_atomic_min_u32 (no-return, STOREcnt).
// ---------------------------------------------------------------------------
__global__ void rs2d_zmin_kernel(const float* __restrict__ flow,
                                 const float* __restrict__ depth,
                                 unsigned int* __restrict__ dlut) {
    const int i = blockIdx.x * blockDim.x + threadIdx.x;
    if (i >= NPIX) return;
    const int b   = i >> HW_SHIFT;
    const int rem = i & HW_MASK;
    const int tloc = warp_target(flow, b, rem);
    const int tgt  = (b << HW_SHIFT) + tloc;
    const float d  = depth[i];
    atomicMin(&dlut[tgt], __float_as_uint(d));
}

// ---------------------------------------------------------------------------
// Pass 2: keep-test against the z-buffer winner, then scatter-add the 3
// channels straight into the planar (B,C,H,W) output and bump the count.
// obj channels are prefetched (global_prefetch_b8 on gfx1250) so the payload
// fetch overlaps the dependent dlut load. All scatter state (~84 MB) is
// resident in the 192 MB L2, so these atomics never round-trip to HBM.
// ---------------------------------------------------------------------------
__global__ void rs2d_splat_kernel(const float* __restrict__ obj,
                                  const float* __restrict__ flow,
                                  const float* __restrict__ depth,
                                  const unsigned int* __restrict__ dlut,
                                  float* __restrict__ cnt,
                                  float* __restrict__ out) {
    const int i = blockIdx.x * blockDim.x + threadIdx.x;
    if (i >= NPIX) return;
    const int b   = i >> HW_SHIFT;
    const int rem = i & HW_MASK;

    const size_t sbase = (size_t)b * (size_t)(CC * HWP) + (size_t)rem;
    __builtin_prefetch(&obj[sbase], 0, 0);           // -> global_prefetch_b8
    __builtin_prefetch(&obj[sbase + HWP], 0, 0);
    __builtin_prefetch(&obj[sbase + 2 * HWP], 0, 0);

    const int tloc = warp_target(flow, b, rem);
    const int tgt  = (b << HW_SHIFT) + tloc;

    const float d    = depth[i];
    const float dmin = __uint_as_float(dlut[tgt]);
    if (d <= dmin + SAME_RANGE) {
        const size_t tbase = (size_t)b * (size_t)(CC * HWP) + (size_t)tloc;
        atomicAdd(&cnt[tgt], 1.0f);
        atomicAdd(&out[tbase],            obj[sbase]);
        atomicAdd(&out[tbase + HWP],      obj[sbase + HWP]);
        atomicAdd(&out[tbase + 2 * HWP],  obj[sbase + 2 * HWP]);
    }
}

// ---------------------------------------------------------------------------
// Pass 3: normalize sums by count. Count==0 targets already hold 0.0f from
// the init pass, matching jnp.where(cnts>0, sums/max(cnts,1), 0).
// ---------------------------------------------------------------------------
__global__ void rs2d_norm_kernel(const float* __restrict__ cnt,
                                 float* __restrict__ out) {
    const int i = blockIdx.x * blockDim.x + threadIdx.x;
    if (i >= NPIX) return;
    const float c = cnt[i];
    if (c > 0.0f) {
        const int b   = i >> HW_SHIFT;
        const int loc = i & HW_MASK;
        const size_t tbase = (size_t)b * (size_t)(CC * HWP) + (size_t)loc;
        const float inv = 1.0f / c;   // cnt >= 1 here, so max(cnt,1) == cnt
        out[tbase]           *= inv;
        out[tbase + HWP]     *= inv;
        out[tbase + 2 * HWP] *= inv;
    }
}

// ---------------------------------------------------------------------------
// Host launcher. Inputs per setup_inputs(): [0]=obj f32, [1]=flow f32,
// [2]=depth f32. Output: f32, B*C*H*W planar. Workspace: dlut (N u32) +
// cnt (N f32) = 33.5 MB. Stream ordering provides the inter-pass barriers
// the segment_min / segment_sum semantics require.
// ---------------------------------------------------------------------------
extern "C" void kernel_launch(void* const* d_in, const int* in_sizes, int n_in,
                              void* d_out, int out_size, void* d_ws, size_t ws_size,
                              hipStream_t stream) {
    const float* obj   = (const float*)d_in[0];
    const float* flow  = (const float*)d_in[1];
    const float* depth = (const float*)d_in[2];
    float* out = (float*)d_out;

    unsigned int* dlut = (unsigned int*)d_ws;
    float* cnt = (float*)((char*)d_ws + (size_t)NPIX * sizeof(unsigned int));

    const int threads = 256;  // 8 wave32 waves per workgroup
    const int N4      = NPIX / 4;
    const int gridIni = (CC * N4 + threads - 1) / threads;
    const int gridN   = (NPIX + threads - 1) / threads;

    rs2d_init_kernel<<<gridIni, threads, 0, stream>>>(
        (float4*)out, (uint4*)dlut, (float4*)cnt);
    rs2d_zmin_kernel<<<gridN, threads, 0, stream>>>(flow, depth, dlut);
    rs2d_splat_kernel<<<gridN, threads, 0, stream>>>(obj, flow, depth, dlut, cnt, out);
    rs2d_norm_kernel<<<gridN, threads, 0, stream>>>(cnt, out);
}